// CausalAttention_26259430048036
// MI455X (gfx1250) — compile-verified
//
#include <hip/hip_runtime.h>
#include <math.h>

// CDNA5 / gfx1250, wave32. FP32 matrix path: V_WMMA_F32_16X16X4_F32.
// Staging uses GLOBAL_LOAD_ASYNC_TO_LDS_B64 (ASYNCcnt) where LDS dst is
// contiguous per-lane.
typedef __attribute__((ext_vector_type(2))) float v2f;
typedef __attribute__((ext_vector_type(8))) float v8f;

constexpr int kB  = 2;
constexpr int kT  = 2048;
constexpr int kE  = 1024;
constexpr int kH  = 16;
constexpr int kHD = 64;

__device__ __forceinline__ v8f wmma_f32(v2f a, v2f b, v8f c) {
  // 8 args: (neg_a, A, neg_b, B, c_mod, C, reuse_a, reuse_b)
  return __builtin_amdgcn_wmma_f32_16x16x4_f32(false, a, false, b, (short)0, c,
                                               false, false);
}

// Async global->LDS copy, 8 bytes per lane. VDST holds the LDS byte address
// (low 32 bits of a generic pointer to __shared__ == LDS offset, ISA 10.2).
__device__ __forceinline__ void async_copy_b64(unsigned lds_addr,
                                               const void* gaddr) {
  asm volatile("global_load_async_to_lds_b64 %0, %1, off"
               :: "v"(lds_addr), "v"(gaddr)
               : "memory");
}
__device__ __forceinline__ void async_wait0() {
  asm volatile("s_wait_asynccnt 0x0" ::: "memory");
}

// ---------------------------------------------------------------------------
// C[M,N] = A[M,K] @ W[N,K]^T + bias[N]
// Block: 256 threads = 8 waves. Block tile 128(M) x 128(N), K staged 16 wide.
// Waves 4(M) x 2(N); each wave owns 32x64 = 8 WMMA accumulators.
// Both tiles are row-major with word stride 18 (odd*2: fragment reads are
// single aligned b64, bank-conflict-free). The B fragment (W[n][k],W[n][k+1])
// is contiguous in the row-major W tile, so staging is a pure async copy.
// Assumes M%128==0, N%128==0, K%16==0 (true here: 4096/3072/1024/1024).
// ---------------------------------------------------------------------------
__global__ __launch_bounds__(256) void gemm_bias_kernel(
    const float* __restrict__ A, const float* __restrict__ W,
    const float* __restrict__ bias, float* __restrict__ C,
    int M, int N, int K)
{
  __shared__ float AsF[128 * 18];   // 9216 B: AsF[m*18 + k]
  __shared__ float WsF[128 * 18];   // 9216 B: WsF[n*18 + k]

  const int tid   = threadIdx.x;
  const int lane  = tid & 31;
  const int wid   = tid >> 5;
  const int waveM = wid & 3;      // 0..3  -> 32-row strips
  const int waveN = wid >> 2;     // 0..1  -> 64-col strips
  const int m0    = blockIdx.y * 128;
  const int n0    = blockIdx.x * 128;

  v8f acc0[4], acc1[4];
  for (int nt = 0; nt < 4; ++nt)
    for (int r = 0; r < 8; ++r) { acc0[nt][r] = 0.0f; acc1[nt][r] = 0.0f; }

  const int mLane = lane & 15;          // fragment row/col within 16
  const int kHalf = (lane >> 4) << 1;   // lanes 16-31 hold K+2,K+3

  const int sr = tid >> 1;              // staged row 0..127
  const int sk = (tid & 1) << 3;        // k sub-offset 0 or 8
  const unsigned ldsA = (unsigned)(uintptr_t)&AsF[sr * 18 + sk];
  const unsigned ldsW = (unsigned)(uintptr_t)&WsF[sr * 18 + sk];

  for (int k0 = 0; k0 < K; k0 += 16) {
    __syncthreads();   // previous tile consumers done
    {   // async stage A: 8 contiguous floats/thread
      const float* ap = A + (size_t)(m0 + sr) * K + k0 + sk;
      async_copy_b64(ldsA,      ap);
      async_copy_b64(ldsA + 8,  ap + 2);
      async_copy_b64(ldsA + 16, ap + 4);
      async_copy_b64(ldsA + 24, ap + 6);
    }
    {   // async stage W: 8 contiguous floats/thread
      const float* wp = W + (size_t)(n0 + sr) * K + k0 + sk;
      async_copy_b64(ldsW,      wp);
      async_copy_b64(ldsW + 8,  wp + 2);
      async_copy_b64(ldsW + 16, wp + 4);
      async_copy_b64(ldsW + 24, wp + 6);
    }
    async_wait0();
    __syncthreads();

#pragma unroll
    for (int kk = 0; kk < 16; kk += 4) {
      const int kA = kk + kHalf;   // even
      const v2f a0 = *(const v2f*)&AsF[(waveM * 32 + mLane) * 18 + kA];
      const v2f a1 = *(const v2f*)&AsF[(waveM * 32 + 16 + mLane) * 18 + kA];
#pragma unroll
      for (int nt = 0; nt < 4; ++nt) {
        const int nc = waveN * 64 + nt * 16 + mLane;
        const v2f b = *(const v2f*)&WsF[nc * 18 + kA];
        acc0[nt] = wmma_f32(a0, b, acc0[nt]);
        acc1[nt] = wmma_f32(a1, b, acc1[nt]);
      }
    }
  }

  // Epilogue: C layout — lane half picks M block of 8, vgpr r picks row.
  const int rowHalf = (lane >> 4) << 3;
#pragma unroll
  for (int nt = 0; nt < 4; ++nt) {
    const int nc = n0 + waveN * 64 + nt * 16 + mLane;
    const float bv = bias[nc];
    const int mB0 = m0 + waveM * 32 + rowHalf;
#pragma unroll
    for (int r = 0; r < 8; ++r) {
      C[(size_t)(mB0 + r) * N + nc]      = acc0[nt][r] + bv;
      C[(size_t)(mB0 + 16 + r) * N + nc] = acc1[nt][r] + bv;
    }
  }
}

// ---------------------------------------------------------------------------
// Flash attention over qkv buffer [B, T, 3E] (heads packed in E).
// Grid: (T/64, H, B); block = 128 threads = 4 waves; each wave owns 16 query
// rows. Q/K tiles staged with async copies; V stored TRANSPOSED in LDS
// (manual scatter) so the O=P*V B-fragment is one contiguous b64.
// All LDS arrays use word stride 82 (41 pairs; conflict-free fragment reads).
// ---------------------------------------------------------------------------
__global__ __launch_bounds__(128) void attention_kernel(
    const float* __restrict__ qkv, float* __restrict__ y)
{
  __shared__ float QsF[64 * 82];   // Qs[row*82 + d]
  __shared__ float KsF[64 * 82];   // Ks[key*82 + d]
  __shared__ float VtF[64 * 82];   // Vt[d*82 + key]   (transposed!)
  __shared__ float PsF[64 * 82];   // Ps[row*82 + key] (wave-private rows)

  const int tid  = threadIdx.x;
  const int lane = tid & 31;
  const int wid  = tid >> 5;
  const int h    = blockIdx.y;
  const int b    = blockIdx.z;
  const int q0   = blockIdx.x * 64;

  const size_t rowStride = 3 * kE;
  const float* qBase = qkv + (size_t)(b * kT) * rowStride + h * kHD;
  const float* kBase = qBase + kE;
  const float* vBase = qBase + 2 * kE;

  // Async-stage Q tile (64 x 64): 16B per thread-iteration as 2x b64
  for (int i = tid; i < 64 * 16; i += 128) {
    const int row = i >> 4;
    const int c   = (i & 15) << 2;
    const unsigned qdst = (unsigned)(uintptr_t)&QsF[row * 82 + c];
    const float* qsrc = qBase + (size_t)(q0 + row) * rowStride + c;
    async_copy_b64(qdst,     qsrc);
    async_copy_b64(qdst + 8, qsrc + 2);
  }

  float mSt[8], lSt[8];
  v8f o[4];
  for (int r = 0; r < 8; ++r) { mSt[r] = -INFINITY; lSt[r] = 0.0f; }
  for (int dt = 0; dt < 4; ++dt)
    for (int r = 0; r < 8; ++r) o[dt][r] = 0.0f;

  const int mLane   = lane & 15;
  const int kHalf   = (lane >> 4) << 1;
  const int rowHalf = (lane >> 4) << 3;
  const float scale = 0.125f;   // 1/sqrt(64)

  for (int kb = 0; kb <= (int)blockIdx.x; ++kb) {
    __syncthreads();   // previous K/V consumers done
    for (int i = tid; i < 64 * 16; i += 128) {
      const int row = i >> 4;
      const int c   = (i & 15) << 2;
      // K: async contiguous copy
      const unsigned kdst = (unsigned)(uintptr_t)&KsF[row * 82 + c];
      const float* ksrc = kBase + (size_t)(kb * 64 + row) * rowStride + c;
      async_copy_b64(kdst,     ksrc);
      async_copy_b64(kdst + 8, ksrc + 2);
      // V: manual transpose into LDS
      const float4 vv = *(const float4*)(vBase + (size_t)(kb * 64 + row) * rowStride + c);
      VtF[(c + 0) * 82 + row] = vv.x;
      VtF[(c + 1) * 82 + row] = vv.y;
      VtF[(c + 2) * 82 + row] = vv.z;
      VtF[(c + 3) * 82 + row] = vv.w;
    }
    async_wait0();     // covers this wave's K asyncs (and Q on first iter)
    __syncthreads();

    // S = Q * K^T  (16 queries x 64 keys per wave)
    v8f s[4];
    for (int nt = 0; nt < 4; ++nt)
      for (int r = 0; r < 8; ++r) s[nt][r] = 0.0f;
#pragma unroll
    for (int kk = 0; kk < kHD; kk += 4) {
      const int kA = kk + kHalf;   // even
      const v2f a = *(const v2f*)&QsF[(wid * 16 + mLane) * 82 + kA];
#pragma unroll
      for (int nt = 0; nt < 4; ++nt) {
        const v2f bb = *(const v2f*)&KsF[(nt * 16 + mLane) * 82 + kA];
        s[nt] = wmma_f32(a, bb, s[nt]);
      }
    }

    // Scale + causal mask + online softmax. Row m lives in one 16-lane half.
    const int qRow0 = q0 + wid * 16 + rowHalf;
    float rmax[8];
    for (int r = 0; r < 8; ++r) rmax[r] = mSt[r];
    for (int nt = 0; nt < 4; ++nt) {
      const int kc = kb * 64 + nt * 16 + mLane;
      for (int r = 0; r < 8; ++r) {
        float sv = s[nt][r] * scale;
        sv = (kc > qRow0 + r) ? -INFINITY : sv;
        s[nt][r] = sv;
        rmax[r] = fmaxf(rmax[r], sv);
      }
    }
    for (int r = 0; r < 8; ++r)
      for (int off = 1; off < 16; off <<= 1)
        rmax[r] = fmaxf(rmax[r], __shfl_xor(rmax[r], off, 16));

    float rsum[8];
    for (int r = 0; r < 8; ++r) rsum[r] = 0.0f;
    for (int nt = 0; nt < 4; ++nt)
      for (int r = 0; r < 8; ++r) {
        const float p = __expf(s[nt][r] - rmax[r]);
        s[nt][r] = p;
        rsum[r] += p;
      }
    for (int r = 0; r < 8; ++r) {
      for (int off = 1; off < 16; off <<= 1)
        rsum[r] += __shfl_xor(rsum[r], off, 16);
      const float alpha = __expf(mSt[r] - rmax[r]);
      lSt[r] = lSt[r] * alpha + rsum[r];
      mSt[r] = rmax[r];
      for (int dt = 0; dt < 4; ++dt) o[dt][r] *= alpha;
    }

    // P: C-layout -> A-layout via wave-private LDS rows [wid*16, wid*16+16)
    for (int nt = 0; nt < 4; ++nt)
      for (int r = 0; r < 8; ++r)
        PsF[(wid * 16 + rowHalf + r) * 82 + nt * 16 + mLane] = s[nt][r];

    // O += P * V  (V transposed in LDS: pair along keys is contiguous)
#pragma unroll
    for (int kk = 0; kk < 64; kk += 4) {
      const int kA = kk + kHalf;   // even (key pair base)
      const v2f a = *(const v2f*)&PsF[(wid * 16 + mLane) * 82 + kA];
#pragma unroll
      for (int dt = 0; dt < 4; ++dt) {
        const v2f bb = *(const v2f*)&VtF[(dt * 16 + mLane) * 82 + kA];
        o[dt] = wmma_f32(a, bb, o[dt]);
      }
    }
  }

  // Finalize: divide by l, write y[B,T,E]
  const int outRow0 = q0 + wid * 16 + rowHalf;
  for (int dt = 0; dt < 4; ++dt)
    for (int r = 0; r < 8; ++r) {
      const float val = o[dt][r] / lSt[r];
      y[(size_t)(b * kT + outRow0 + r) * kE + h * kHD + dt * 16 + mLane] = val;
    }
}

// ---------------------------------------------------------------------------
extern "C" void kernel_launch(void* const* d_in, const int* in_sizes, int n_in,
                              void* d_out, int out_size, void* d_ws, size_t ws_size,
                              hipStream_t stream)
{
  const float* x     = (const float*)d_in[0];  // [B,T,E]
  const float* qkv_w = (const float*)d_in[1];  // [3E,E]
  const float* qkv_b = (const float*)d_in[2];  // [3E]
  const float* o_w   = (const float*)d_in[3];  // [E,E]
  const float* o_b   = (const float*)d_in[4];  // [E]
  float* out = (float*)d_out;                  // [B,T,E]

  // Workspace: qkv [4096,3072] (50.3 MB) + y [4096,1024] (16.8 MB)
  float* qkv = (float*)d_ws;
  float* yb  = qkv + (size_t)kB * kT * 3 * kE;

  const int M = kB * kT;   // 4096

  // 1) qkv = x @ qkv_w^T + qkv_b
  gemm_bias_kernel<<<dim3((3 * kE) / 128, M / 128), 256, 0, stream>>>(
      x, qkv_w, qkv_b, qkv, M, 3 * kE, kE);

  // 2) flash attention -> y
  attention_kernel<<<dim3(kT / 64, kH, kB), 128, 0, stream>>>(qkv, yb);

  // 3) out = y @ o_w^T + o_b
  gemm_bias_kernel<<<dim3(kE / 128, M / 128), 256, 0, stream>>>(
      yb, o_w, o_b, out, M, kE, kE);
}